// CosineSimCodebook_58531814310488
// MI455X (gfx1250) — compile-verified
//
#include <hip/hip_runtime.h>
#include <hip/hip_bf16.h>

typedef __attribute__((ext_vector_type(2))) float v2f;
typedef __attribute__((ext_vector_type(8))) float v8f;

#define CBOOK 4096
#define DDIM  512
#define NTOK  32768          // B*N = 8*4096
#define DECAY 0.8f
#define OMDECAY 0.2f
#define EPSV  1e-5f

// ---- kernel 1 config ----
#define BLK_M 128            // rows of X per workgroup (4 waves * 32 rows)
#define COLT  16             // dist columns per iteration (one WMMA N-tile)
#define XS    516            // padded LDS stride (516 mod 64 = 4 -> conflict-free b64)
#define ES    516

// LDS-only split barrier: order DS traffic, do NOT wait for the in-flight
// non-temporal global stores (plain __syncthreads emits s_wait_storecnt 0x0).
#define LDS_BARRIER() asm volatile(                                            \
    "s_wait_dscnt 0x0\n\t"                                                     \
    "s_barrier_signal -1\n\t"                                                  \
    "s_barrier_wait -1" ::: "memory")

// ============================================================================
// Kernel 1: dist = X @ E^T via V_WMMA_F32_16X16X4_F32, fused row-argmax.
// grid = NTOK/BLK_M blocks, 128 threads (4 waves, 32 rows each: one B fragment
// feeds two WMMAs with independent accumulators -> 3 b64 DS loads / 2 WMMAs).
// ============================================================================
__global__ void __launch_bounds__(128)
dist_argmax_kernel(const float* __restrict__ X, const float* __restrict__ E,
                   float* __restrict__ dist, float* __restrict__ ind_f,
                   int* __restrict__ ind_i)
{
    extern __shared__ float smem[];
    float* sX = smem;                 // [BLK_M][XS]
    float* sE = smem + BLK_M * XS;    // [COLT][ES]

    const int tid    = threadIdx.x;
    const int wave   = tid >> 5;
    const int lane   = tid & 31;
    const int lane16 = lane & 15;
    const int half   = lane >> 4;
    const int m0     = blockIdx.x * BLK_M;
    const int wr0    = wave * 32;     // this wave's 32 rows within the block

    // ---- stage X tile (128 x 512 fp32) into LDS, float4 coalesced ----
    {
        const float4* Xv = (const float4*)(X + (long)m0 * DDIM);
        for (int i = tid; i < BLK_M * (DDIM / 4); i += 128) {
            int r = i >> 7;            // / (DDIM/4)
            int c = (i & 127) * 4;     // % (DDIM/4) * 4
            float4 v = Xv[(long)r * (DDIM / 4) + (c >> 2)];
            *(float4*)&sX[r * XS + c] = v;
        }
    }
    LDS_BARRIER();

    // running argmax state: acc slot i <-> row (wr0 [+16] + half*8 + i)
    v8f rmax0, rmax1;
    int ridx0[8], ridx1[8];
#pragma unroll
    for (int i = 0; i < 8; ++i) {
        rmax0[i] = -3.4e38f; ridx0[i] = 0;
        rmax1[i] = -3.4e38f; ridx1[i] = 0;
    }

    const float* aBase0 = &sX[(wr0 + lane16) * XS + half * 2];
    const float* aBase1 = aBase0 + 16 * XS;

    for (int ct = 0; ct < CBOOK / COLT; ++ct) {
        const int c0 = ct * COLT;

        // ---- stage 16 codebook rows (B-matrix tile) into LDS ----
        {
            const float4* Ev = (const float4*)(E + (long)c0 * DDIM);
            for (int i = tid; i < COLT * (DDIM / 4); i += 128) {
                int r = i >> 7;
                int c = (i & 127) * 4;
                float4 v = Ev[(long)r * (DDIM / 4) + (c >> 2)];
                *(float4*)&sE[r * ES + c] = v;
            }
        }
        // prefetch next codebook tile into cache while we compute this one
        if (ct + 1 < CBOOK / COLT)
            __builtin_prefetch(&E[(long)(c0 + COLT) * DDIM + tid * 64], 0, 1);
        LDS_BARRIER();

        // ---- two 16x16 dist tiles: 128 k-steps, 1 B frag shared by 2 WMMAs --
        v8f acc0 = {0.f, 0.f, 0.f, 0.f, 0.f, 0.f, 0.f, 0.f};
        v8f acc1 = {0.f, 0.f, 0.f, 0.f, 0.f, 0.f, 0.f, 0.f};
        const float* bBase = &sE[lane16 * ES + half * 2];
#pragma unroll 8
        for (int ks = 0; ks < DDIM / 4; ++ks) {
            v2f a0; a0.x = aBase0[ks * 4]; a0.y = aBase0[ks * 4 + 1];
            v2f a1; a1.x = aBase1[ks * 4]; a1.y = aBase1[ks * 4 + 1];
            v2f b;  b.x  = bBase[ks * 4];  b.y  = bBase[ks * 4 + 1];
            acc0 = __builtin_amdgcn_wmma_f32_16x16x4_f32(
                false, a0, false, b, (short)0, acc0, false, false);
            acc1 = __builtin_amdgcn_wmma_f32_16x16x4_f32(
                false, a1, false, b, (short)0, acc1, false, false);
        }

        // ---- stream dist tiles out (non-temporal) + update running argmax ---
#pragma unroll
        for (int i = 0; i < 8; ++i) {
            int row0 = m0 + wr0 + half * 8 + i;
            __builtin_nontemporal_store(acc0[i],
                &dist[(long)row0 * CBOOK + c0 + lane16]);
            if (acc0[i] > rmax0[i]) { rmax0[i] = acc0[i]; ridx0[i] = c0 + lane16; }
            int row1 = row0 + 16;
            __builtin_nontemporal_store(acc1[i],
                &dist[(long)row1 * CBOOK + c0 + lane16]);
            if (acc1[i] > rmax1[i]) { rmax1[i] = acc1[i]; ridx1[i] = c0 + lane16; }
        }
        LDS_BARRIER();   // before next tile overwrites sE (DS-only wait)
    }

    // ---- reduce argmax across the 16 lanes of each half-wave ----
#pragma unroll
    for (int i = 0; i < 8; ++i) {
        float m0v = rmax0[i]; int id0 = ridx0[i];
        float m1v = rmax1[i]; int id1 = ridx1[i];
#pragma unroll
        for (int off = 1; off < 16; off <<= 1) {
            float om0 = __shfl_xor(m0v, off, 32);
            int   oi0 = __shfl_xor(id0, off, 32);
            if (om0 > m0v || (om0 == m0v && oi0 < id0)) { m0v = om0; id0 = oi0; }
            float om1 = __shfl_xor(m1v, off, 32);
            int   oi1 = __shfl_xor(id1, off, 32);
            if (om1 > m1v || (om1 == m1v && oi1 < id1)) { m1v = om1; id1 = oi1; }
        }
        if (lane16 == 0) {
            int row0 = m0 + wr0 + half * 8 + i;
            ind_i[row0] = id0;
            ind_f[row0] = (float)id0;
            int row1 = row0 + 16;
            ind_i[row1] = id1;
            ind_f[row1] = (float)id1;
        }
    }
}

// ============================================================================
// Kernel 2: quantize gather + embed_sum scatter-add + bins histogram.
// one block per token, 128 threads (4 dims each).
// ============================================================================
__global__ void __launch_bounds__(128)
gather_scatter_kernel(const float* __restrict__ X, const float* __restrict__ E,
                      const int* __restrict__ ind, float* __restrict__ quant,
                      float* __restrict__ embed_sum, int* __restrict__ bins)
{
    const int n   = blockIdx.x;
    const int tid = threadIdx.x;
    const int idx = ind[n];

    const float4* Ev = (const float4*)(E + (long)idx * DDIM);
    const float4* Xv = (const float4*)(X + (long)n * DDIM);
    float4*       Qv = (float4*)(quant + (long)n * DDIM);

    float4 e  = Ev[tid];
    float4 xv = Xv[tid];
    Qv[tid] = e;

    float* es = embed_sum + (long)idx * DDIM + tid * 4;
    atomicAdd(es + 0, xv.x);
    atomicAdd(es + 1, xv.y);
    atomicAdd(es + 2, xv.z);
    atomicAdd(es + 3, xv.w);
    if (tid == 0) atomicAdd(&bins[idx], 1);
}

// ============================================================================
// Kernel 3: new_cluster + laplace smoothing (single block, C=4096).
// ============================================================================
__global__ void __launch_bounds__(256)
cluster_kernel(const float* __restrict__ cluster_size, const int* __restrict__ bins,
               float* __restrict__ new_cluster, float* __restrict__ smoothed)
{
    __shared__ float red[256];
    const int tid = threadIdx.x;
    float nc[CBOOK / 256];
    float lsum = 0.f;
#pragma unroll
    for (int i = 0; i < CBOOK / 256; ++i) {
        int c = tid + i * 256;
        float v = cluster_size[c] * DECAY + (float)bins[c] * OMDECAY;
        nc[i] = v;
        new_cluster[c] = v;
        lsum += v;
    }
    red[tid] = lsum;
    __syncthreads();
    for (int s = 128; s > 0; s >>= 1) {
        if (tid < s) red[tid] += red[tid + s];
        __syncthreads();
    }
    const float tot = red[0];
    const float denom = tot + (float)CBOOK * EPSV;
#pragma unroll
    for (int i = 0; i < CBOOK / 256; ++i) {
        int c = tid + i * 256;
        smoothed[c] = (nc[i] + EPSV) / denom * tot;
    }
}

// ============================================================================
// Kernel 4: new_embed_avg (EMA) + new_embed = l2norm(new_embed_avg/smoothed).
// one block per codebook row.
// ============================================================================
__global__ void __launch_bounds__(256)
embed_update_kernel(const float* __restrict__ embed_avg,
                    const float* __restrict__ embed_sum,
                    const float* __restrict__ smoothed,
                    float* __restrict__ new_embed_avg,
                    float* __restrict__ new_embed)
{
    __shared__ float red[256];
    const int c   = blockIdx.x;
    const int tid = threadIdx.x;
    const float s = smoothed[c];

    float t[2];
    float lsum = 0.f;
#pragma unroll
    for (int i = 0; i < 2; ++i) {
        int d = tid + i * 256;
        float v = embed_avg[(long)c * DDIM + d] * DECAY +
                  embed_sum[(long)c * DDIM + d] * OMDECAY;
        new_embed_avg[(long)c * DDIM + d] = v;
        t[i] = v / s;
        lsum += t[i] * t[i];
    }
    red[tid] = lsum;
    __syncthreads();
    for (int st = 128; st > 0; st >>= 1) {
        if (tid < st) red[tid] += red[tid + st];
        __syncthreads();
    }
    float nrm = fmaxf(sqrtf(red[0]), 1e-12f);
#pragma unroll
    for (int i = 0; i < 2; ++i) {
        int d = tid + i * 256;
        new_embed[(long)c * DDIM + d] = t[i] / nrm;
    }
}

// ============================================================================
extern "C" void kernel_launch(void* const* d_in, const int* in_sizes, int n_in,
                              void* d_out, int out_size, void* d_ws, size_t ws_size,
                              hipStream_t stream)
{
    const float* x            = (const float*)d_in[0];   // (B,N,D)
    const float* embed        = (const float*)d_in[1];   // (H,C,D)
    const float* cluster_size = (const float*)d_in[2];   // (H,C)
    const float* embed_avg    = (const float*)d_in[3];   // (H,C,D)

    // d_out: outputs concatenated flat in return order (all fp32)
    float* out           = (float*)d_out;
    float* quant         = out;                // 8*4096*512   = 16,777,216
    float* ind_f         = out + 16777216L;    // 8*4096       =     32,768
    float* dist          = out + 16809984L;    // 32768*4096   = 134,217,728
    float* new_cluster   = out + 151027712L;   // 4096
    float* new_embed_avg = out + 151031808L;   // 4096*512     =  2,097,152
    float* new_embed     = out + 153128960L;   // 4096*512     =  2,097,152

    // workspace layout
    char*  ws        = (char*)d_ws;
    int*   ind_i     = (int*)ws;               // 32768 ints
    int*   bins      = (int*)(ws + 131072);    // 4096 ints
    float* smoothed  = (float*)(ws + 147456);  // 4096 floats
    float* embed_sum = (float*)(ws + 163840);  // 4096*512 floats (8 MB)

    // zero the accumulators every call (graph-capture-safe async memsets)
    hipMemsetAsync(bins, 0, 4096 * sizeof(int), stream);
    hipMemsetAsync(embed_sum, 0, (size_t)CBOOK * DDIM * sizeof(float), stream);

    const size_t lds_bytes = (size_t)(BLK_M * XS + COLT * ES) * sizeof(float); // 297,216 B
    hipFuncSetAttribute((const void*)dist_argmax_kernel,
                        hipFuncAttributeMaxDynamicSharedMemorySize,
                        (int)lds_bytes);

    dist_argmax_kernel<<<NTOK / BLK_M, 128, lds_bytes, stream>>>(
        x, embed, dist, ind_f, ind_i);
    gather_scatter_kernel<<<NTOK, 128, 0, stream>>>(
        x, embed, ind_i, quant, embed_sum, bins);
    cluster_kernel<<<1, 256, 0, stream>>>(
        cluster_size, bins, new_cluster, smoothed);
    embed_update_kernel<<<CBOOK, 256, 0, stream>>>(
        embed_avg, embed_sum, smoothed, new_embed_avg, new_embed);
}